// GCN_69329362092525
// MI455X (gfx1250) — compile-verified
//
#include <hip/hip_runtime.h>

// CDNA5 (gfx1250) 2-layer GCN.
// Dense transforms use V_WMMA_F32_16X16X4_F32 (wave32, 16x16 f32 tiles),
// full tiles only (no guards in the WMMA path); scalar tail kernels cover
// N % 16 leftovers. Sparse aggregation uses edge-parallel
// global_atomic_add_f32 into L2-resident scratch (h1/agg1 = 12.8MB << 192MB L2).

typedef __attribute__((ext_vector_type(2))) float v2f;
typedef __attribute__((ext_vector_type(8))) float v8f;

// ---------------- init: deg=1 (self loop), agg1=0, out=0 ----------------
__global__ void k_init(float* __restrict__ deg, float* __restrict__ agg1,
                       float* __restrict__ out, int N) {
  int i = blockIdx.x * blockDim.x + threadIdx.x;
  if (i < 16 * N) agg1[i] = 0.0f;
  if (i < 2 * N)  out[i]  = 0.0f;
  if (i < N)      deg[i]  = 1.0f;   // self-loop contribution to in-degree
}

// ---------------- degree accumulation over edges ----------------
__global__ void k_degree(const int* __restrict__ dst, float* __restrict__ deg, int E) {
  int e = blockIdx.x * blockDim.x + threadIdx.x;
  if (e < E) atomicAdd(&deg[dst[e]], 1.0f);
}

// ---------------- dinv = rsqrt(deg) in place ----------------
__global__ void k_dinv(float* __restrict__ deg, int N) {
  int i = blockIdx.x * blockDim.x + threadIdx.x;
  if (i < N) {
    float d = deg[i];
    deg[i] = (d > 0.0f) ? rsqrtf(d) : 0.0f;
  }
}

// ---------------- h1 = x @ W1  (N x 2) @ (2 x 16) via WMMA f32 16x16x4 ----
// Full tiles only: every wave owns 16 valid nodes -> straight-line loads,
// one v_wmma, 8 unconditional coalesced stores per lane.
// A (16x4): lanes 0-15 hold row M=lane, K={0,1} in v[0],v[1]; lanes 16-31
// hold K={2,3} (zero-padded, real K=2). B (4x16) mirrored with N across lanes.
__global__ void k_gemm1(const float* __restrict__ x, const float* __restrict__ W1,
                        float* __restrict__ h1, int ntiles) {
  int wave = (blockIdx.x * blockDim.x + threadIdx.x) >> 5;
  int lane = threadIdx.x & 31;
  if (wave >= ntiles) return;
  int node0 = wave << 4;

  v2f a = {0.0f, 0.0f};
  v2f b = {0.0f, 0.0f};
  if (lane < 16) {
    int node = node0 + lane;
    a.x = x[node * 2 + 0];                  // K=0
    a.y = x[node * 2 + 1];                  // K=1
    b.x = W1[0 * 16 + lane];                // B row K=0, col N=lane
    b.y = W1[1 * 16 + lane];                // B row K=1
  }                                         // lanes 16-31: K=2,3 -> zero pad

  v8f c = {};
  c = __builtin_amdgcn_wmma_f32_16x16x4_f32(
      /*neg_a=*/false, a, /*neg_b=*/false, b,
      /*c_mod=*/(short)0, c, /*reuse_a=*/false, /*reuse_b=*/false);

  // D layout: VGPR j -> rows {node0+j (lanes 0-15), node0+8+j (lanes 16-31)}
  int row0 = node0 + ((lane < 16) ? 0 : 8);
  int col  = lane & 15;
#pragma unroll
  for (int j = 0; j < 8; ++j)
    h1[(row0 + j) * 16 + col] = c[j];
}

// scalar tail for nodes >= ntiles*16 (empty when N % 16 == 0)
__global__ void k_gemm1_tail(const float* __restrict__ x, const float* __restrict__ W1,
                             float* __restrict__ h1, int n0, int N) {
  int tid = blockIdx.x * blockDim.x + threadIdx.x;
  int n = n0 + (tid >> 4);
  if (n >= N) return;
  int f = tid & 15;
  h1[n * 16 + f] = x[n * 2 + 0] * W1[f] + x[n * 2 + 1] * W1[16 + f];
}

// ---------------- layer-1 edge aggregation ----------------
// One edge per 16 consecutive lanes -> h1[src*16 + f] gather is a coalesced
// 64B line; edge-index reads are same-cacheline within the wave.
__global__ void k_agg1(const int* __restrict__ src, const int* __restrict__ dst,
                       const float* __restrict__ dinv, const float* __restrict__ h1,
                       float* __restrict__ agg1, int E) {
  int tid = blockIdx.x * blockDim.x + threadIdx.x;
  if (tid >= E * 16) return;
  int e = tid >> 4;
  int f = tid & 15;
  int s = src[e];
  int d = dst[e];
  float nrm = dinv[s] * dinv[d];
  atomicAdd(&agg1[d * 16 + f], h1[s * 16 + f] * nrm);
}

// ---------------- self-loop + bias + ReLU (in place on agg1 -> r) --------
__global__ void k_relu1(const float* __restrict__ h1, const float* __restrict__ dinv,
                        const float* __restrict__ b1, float* __restrict__ agg1, int N) {
  int tid = blockIdx.x * blockDim.x + threadIdx.x;
  if (tid >= N * 16) return;
  int n = tid >> 4;
  int f = tid & 15;
  float di = dinv[n];
  float v = agg1[tid] + h1[tid] * di * di + b1[f];
  agg1[tid] = fmaxf(v, 0.0f);
}

// ---------------- h2 = r @ W2  (N x 16) @ (16 x 2) via 4 chained WMMAs ----
__global__ void k_gemm2(const float* __restrict__ r, const float* __restrict__ W2,
                        float* __restrict__ h2, int ntiles) {
  int wave = (blockIdx.x * blockDim.x + threadIdx.x) >> 5;
  int lane = threadIdx.x & 31;
  if (wave >= ntiles) return;
  int node0 = wave << 4;

  int m = lane & 15;
  int node = node0 + m;

  v8f c = {};
#pragma unroll
  for (int kk = 0; kk < 4; ++kk) {
    int kbase = kk * 4 + ((lane < 16) ? 0 : 2);   // lane-half K split per ISA
    v2f a, b;
    a.x = r[node * 16 + kbase + 0];
    a.y = r[node * 16 + kbase + 1];
    // B: W2 (16x2 row-major) zero-padded to 16 columns
    b.x = (m < 2) ? W2[(kbase + 0) * 2 + m] : 0.0f;
    b.y = (m < 2) ? W2[(kbase + 1) * 2 + m] : 0.0f;
    c = __builtin_amdgcn_wmma_f32_16x16x4_f32(
        false, a, false, b, (short)0, c, false, false);
  }

  // only columns 0,1 of D are meaningful
  int row0 = node0 + ((lane < 16) ? 0 : 8);
  int col  = lane & 15;
  if (col < 2) {
#pragma unroll
    for (int j = 0; j < 8; ++j)
      h2[(row0 + j) * 2 + col] = c[j];
  }
}

// scalar tail for nodes >= ntiles*16 (empty when N % 16 == 0)
__global__ void k_gemm2_tail(const float* __restrict__ r, const float* __restrict__ W2,
                             float* __restrict__ h2, int n0, int N) {
  int tid = blockIdx.x * blockDim.x + threadIdx.x;
  int n = n0 + (tid >> 1);
  if (n >= N) return;
  int f = tid & 1;
  float acc = 0.0f;
#pragma unroll
  for (int k = 0; k < 16; ++k)
    acc += r[n * 16 + k] * W2[k * 2 + f];
  h2[n * 2 + f] = acc;
}

// ---------------- layer-2 edge aggregation (2 features/edge) --------------
__global__ void k_agg2(const int* __restrict__ src, const int* __restrict__ dst,
                       const float* __restrict__ dinv, const float* __restrict__ h2,
                       float* __restrict__ out, int E) {
  int tid = blockIdx.x * blockDim.x + threadIdx.x;
  if (tid >= 2 * E) return;
  int e = tid >> 1;
  int f = tid & 1;
  int s = src[e];
  int d = dst[e];
  float nrm = dinv[s] * dinv[d];
  atomicAdd(&out[d * 2 + f], h2[s * 2 + f] * nrm);
}

// ---------------- final: self-loop + bias ----------------
__global__ void k_final(const float* __restrict__ h2, const float* __restrict__ dinv,
                        const float* __restrict__ b2, float* __restrict__ out, int N) {
  int tid = blockIdx.x * blockDim.x + threadIdx.x;
  if (tid >= 2 * N) return;
  int n = tid >> 1;
  int f = tid & 1;
  float di = dinv[n];
  out[tid] += h2[tid] * di * di + b2[f];
}

extern "C" void kernel_launch(void* const* d_in, const int* in_sizes, int n_in,
                              void* d_out, int out_size, void* d_ws, size_t ws_size,
                              hipStream_t stream) {
  const float* x  = (const float*)d_in[0];   // [N,2]
  const int*   ei = (const int*)d_in[1];     // [2,E]
  const float* W1 = (const float*)d_in[2];   // [2,16]
  const float* b1 = (const float*)d_in[3];   // [16]
  const float* W2 = (const float*)d_in[4];   // [16,2]
  const float* b2 = (const float*)d_in[5];   // [2]
  float* out = (float*)d_out;                // [N,2]

  const int N = in_sizes[0] / 2;
  const int E = in_sizes[1] / 2;
  const int* src = ei;       // edge_index[0]
  const int* dst = ei + E;   // edge_index[1]

  // workspace layout (floats): dinv[N] | h1[16N] | agg1/r[16N] | h2[2N]
  float* ws   = (float*)d_ws;
  float* dinv = ws;
  float* h1   = ws + (size_t)N;
  float* agg1 = ws + (size_t)17 * N;
  float* h2   = ws + (size_t)33 * N;

  const int B = 256;            // 8 wave32 waves per block
  const int ntiles = N / 16;    // full WMMA tiles
  const int n0 = ntiles * 16;   // first tail node
  const int ntail = N - n0;

  k_init  <<<(16 * N + B - 1) / B, B, 0, stream>>>(dinv, agg1, out, N);
  k_degree<<<(E + B - 1) / B,      B, 0, stream>>>(dst, dinv, E);
  k_dinv  <<<(N + B - 1) / B,      B, 0, stream>>>(dinv, N);

  k_gemm1 <<<(ntiles * 32 + B - 1) / B, B, 0, stream>>>(x, W1, h1, ntiles);
  if (ntail > 0)
    k_gemm1_tail<<<(ntail * 16 + B - 1) / B, B, 0, stream>>>(x, W1, h1, n0, N);

  k_agg1  <<<(E * 16 + B - 1) / B, B, 0, stream>>>(src, dst, dinv, h1, agg1, E);
  k_relu1 <<<(16 * N + B - 1) / B, B, 0, stream>>>(h1, dinv, b1, agg1, N);

  k_gemm2 <<<(ntiles * 32 + B - 1) / B, B, 0, stream>>>(agg1, W2, h2, ntiles);
  if (ntail > 0)
    k_gemm2_tail<<<(ntail * 2 + B - 1) / B, B, 0, stream>>>(agg1, W2, h2, n0, N);

  k_agg2  <<<(2 * E + B - 1) / B,  B, 0, stream>>>(src, dst, dinv, h2, out, E);
  k_final <<<(2 * N + B - 1) / B,  B, 0, stream>>>(h2, dinv, b2, out, N);
}